// FeatureWarperSoftsplat_13975823581673
// MI455X (gfx1250) — compile-verified
//
#include <hip/hip_runtime.h>
#include <math.h>

// Softmax splatting (softsplat, strMode='soft', metric == 1) for MI455X.
// Scatter-add workload: memory/atomic bound. Atomic count (268M f32 RMW) is
// the irreducible cost of this formulation; everything is arranged to keep
// the 258MB accumulator L2-resident (NT streaming reads), keep the atomic
// pipes saturated (branch-free clause'd inner loop), and hide the 256KB-
// strided feat stream (gfx1250 global_prefetch, deep unroll). WMMA does not
// apply (scatter has no matmul structure).

namespace {

constexpr int B = 8, C = 128, H = 256, W = 256;
constexpr int HW = H * W;                          // 65536 = 2^16
constexpr float kE = 2.71828182845904523536f;      // exp(metric), metric == 1

typedef float v4f __attribute__((ext_vector_type(4)));

// One thread per source pixel. Bilinear footprint computed once: indices are
// clamped in-bounds and weights of out-of-bounds corners are zeroed (exactly
// the reference's clip + masked-weight formulation), so the channel loop is a
// straight-line sequence of 1 NT load + 4 muls + 4 non-returning atomics with
// no exec-mask churn.
__global__ __launch_bounds__(256) void splat_kernel(const float* __restrict__ feat,
                                                    const float* __restrict__ flow,
                                                    float* __restrict__ out,
                                                    float* __restrict__ wacc)
{
    const int pix = blockIdx.x * 256 + threadIdx.x;    // HW divisible by 256,
    const int b   = pix >> 16;                         // so blocks never straddle b
    const int hw  = pix & (HW - 1);
    const int y   = hw >> 8;                           // W = 256
    const int x   = hw & (W - 1);

    // Flow is streamed once -> non-temporal so it doesn't evict accumulator lines.
    const size_t flo = (size_t)b * 2 * HW + hw;
    const float fx = __builtin_nontemporal_load(flow + flo);
    const float fy = __builtin_nontemporal_load(flow + flo + HW);

    const float tx  = (float)x + fx;
    const float ty  = (float)y + fy;
    const float x0f = floorf(tx);
    const float y0f = floorf(ty);
    const int   x0  = (int)x0f;
    const int   y0  = (int)y0f;
    const float wx1 = tx - x0f, wx0 = 1.0f - wx1;
    const float wy1 = ty - y0f, wy0 = 1.0f - wy1;

    const bool vx0 = (x0 >= 0)  && (x0 < W);
    const bool vx1 = (x0 >= -1) && (x0 < W - 1);
    const bool vy0 = (y0 >= 0)  && (y0 < H);
    const bool vy1 = (y0 >= -1) && (y0 < H - 1);

    // Masked weights (metric factor e folded in); v_cndmask, loop-invariant.
    const float w0 = (vy0 && vx0) ? wy0 * wx0 * kE : 0.0f;
    const float w1 = (vy0 && vx1) ? wy0 * wx1 * kE : 0.0f;
    const float w2 = (vy1 && vx0) ? wy1 * wx0 * kE : 0.0f;
    const float w3 = (vy1 && vx1) ? wy1 * wx1 * kE : 0.0f;

    // All weights zero (fully OOB or degenerate) -> contributes nothing.
    if (w0 + w1 + w2 + w3 == 0.0f) return;

    // Clamped target indices (v_med3_i32); adds of 0.0 to clamped locations
    // are value-preserving, matching the reference's clip+mask semantics.
    const int xc0 = min(max(x0, 0), W - 1);
    const int xc1 = min(max(x0 + 1, 0), W - 1);
    const int yc0 = min(max(y0, 0), H - 1);
    const int yc1 = min(max(y0 + 1, 0), H - 1);
    const int i0 = yc0 * W + xc0;
    const int i1 = yc0 * W + xc1;
    const int i2 = yc1 * W + xc0;
    const int i3 = yc1 * W + xc1;

    // Weight channel (the "129th channel" of the reference).
    float* wp = wacc + (size_t)b * HW;
    atomicAdd(wp + i0, w0);
    atomicAdd(wp + i1, w1);
    atomicAdd(wp + i2, w2);
    atomicAdd(wp + i3, w3);

    const float* fp = feat + (size_t)b * C * HW + hw;
    float*       op = out  + (size_t)b * C * HW;

#pragma unroll 8
    for (int c = 0; c < C; ++c) {
        // gfx1250 global_prefetch 8 channels (2MB) ahead of the strided stream.
        if (c + 8 < C) __builtin_prefetch(fp + (size_t)(c + 8) * HW, 0, 0);
        // feat is used exactly once -> non-temporal read, keep L2 for atomics.
        const float f = __builtin_nontemporal_load(fp + (size_t)c * HW);
        float* dst = op + (size_t)c * HW;
        atomicAdd(dst + i0, f * w0);
        atomicAdd(dst + i1, f * w1);
        atomicAdd(dst + i2, f * w2);
        atomicAdd(dst + i3, f * w3);
    }
}

// out[b,c,hw] /= (wacc[b,hw] == 0 ? 1 : wacc[b,hw]).
// Each thread handles 4 channel planes x float4 at one hw -> the wacc guard
// vector is loaded once and reused 4x; accesses stay 16B-coalesced.
__global__ __launch_bounds__(256) void norm_kernel(float* __restrict__ out,
                                                   const float* __restrict__ wacc)
{
    const size_t t   = (size_t)blockIdx.x * 256 + threadIdx.x; // B*(C/4)*(HW/4) threads
    const int    hw4 = (int)(t & (HW / 4 - 1)) * 4;            // HW/4 = 16384
    const int    cg  = (int)((t >> 14) & (C / 4 - 1)) * 4;     // channel group base
    const int    b   = (int)(t >> 19);                         // (HW/4)*(C/4) = 2^19

    v4f n = *(const v4f*)(wacc + (size_t)b * HW + hw4);        // 2MB, L2-resident
    n.x = (n.x == 0.0f) ? 1.0f : n.x;
    n.y = (n.y == 0.0f) ? 1.0f : n.y;
    n.z = (n.z == 0.0f) ? 1.0f : n.z;
    n.w = (n.w == 0.0f) ? 1.0f : n.w;

    float* base = out + (size_t)b * C * HW + (size_t)cg * HW + hw4;
#pragma unroll
    for (int k = 0; k < 4; ++k) {
        float* p = base + (size_t)k * HW;
        v4f o = *(const v4f*)p;              // hot in L2 from atomics
        o = o / n;
        // Final result: stream out without polluting the cache.
        __builtin_nontemporal_store(o, (v4f*)p);
    }
}

} // namespace

extern "C" void kernel_launch(void* const* d_in, const int* in_sizes, int n_in,
                              void* d_out, int out_size, void* d_ws, size_t ws_size,
                              hipStream_t stream) {
    (void)in_sizes; (void)n_in; (void)ws_size;

    const float* feat = (const float*)d_in[0];   // [B,C,H,W] fp32
    const float* flow = (const float*)d_in[1];   // [B,2,H,W] fp32
    float*       out  = (float*)d_out;           // [B,C,H,W] fp32
    float*       wacc = (float*)d_ws;            // [B,H,W]   fp32 (2MB scratch)

    // Accumulators must start at zero every call (graph replays included).
    hipMemsetAsync(out,  0, (size_t)out_size * sizeof(float), stream);
    hipMemsetAsync(wacc, 0, (size_t)B * HW * sizeof(float), stream);

    // Splat: 8*65536 pixels / 256 = 2048 blocks (8 wave32 each).
    splat_kernel<<<(B * HW) / 256, 256, 0, stream>>>(feat, flow, out, wacc);

    // Normalize: 2^26 elements / 16 per thread / 256 per block = 16384 blocks.
    norm_kernel<<<(B * (size_t)C * HW) / 16 / 256, 256, 0, stream>>>(out, wacc);
}